// UniGraph2_21698174779636
// MI455X (gfx1250) — compile-verified
//
#include <hip/hip_runtime.h>
#include <math.h>

#define NN 384
#define HH 768
#define EE 12288
#define NEXPC 8
#define HEADSC 4
#define LC 3
#define NMC 38
#define FDIM (HEADSC * HH)   // 3072
#define HM_PAD 48            // 38 masked rows padded to multiple of 16

typedef __attribute__((ext_vector_type(2))) float v2f;
typedef __attribute__((ext_vector_type(8))) float v8f;

#if defined(__gfx1250__) && __has_builtin(__builtin_amdgcn_wmma_f32_16x16x4_f32)
#define HAVE_WMMA_F32 1
#endif

// ---------------- helpers ----------------
__device__ inline float wred(float v) {
  v += __shfl_xor(v, 16, 32);
  v += __shfl_xor(v, 8, 32);
  v += __shfl_xor(v, 4, 32);
  v += __shfl_xor(v, 2, 32);
  v += __shfl_xor(v, 1, 32);
  return v;
}

__device__ inline float gelu_f(float x) {
  return 0.5f * x * (1.0f + erff(x * 0.7071067811865476f));
}

__device__ inline unsigned fenc(float f) {
  unsigned b = __float_as_uint(f);
  return (b & 0x80000000u) ? ~b : (b | 0x80000000u);
}
__device__ inline float fdec(unsigned u) {
  unsigned b = (u & 0x80000000u) ? (u & 0x7FFFFFFFu) : ~u;
  return __uint_as_float(b);
}

// ---------------- generic f32 WMMA GEMM ----------------
// C[M,N] = A[M,K] @ B[K,N] (+ bias[n]); if rowscale: C[m,n] += rowscale[m*rs_stride]*val
// One wave per 16x32 tile. M%16==0, N%32==0, K%4==0 (guaranteed by problem sizes).
__global__ __launch_bounds__(32) void gemm_wmma_f32(
    const float* __restrict__ A, const float* __restrict__ B,
    const float* __restrict__ bias, float* __restrict__ C,
    int Nn, int K, const float* __restrict__ rowscale, int rs_stride) {
  const int lane = threadIdx.x;
  const int m0 = blockIdx.y * 16;
  const int n0 = blockIdx.x * 32;
  const int mrow = lane & 15;
  const int kb = (lane >> 4) << 1;  // 0 or 2

  v8f acc0 = {}; v8f acc1 = {};
  const float* Arow = A + (size_t)(m0 + mrow) * K;
#if HAVE_WMMA_F32
  for (int k = 0; k < K; k += 4) {
    v2f a; a.x = Arow[k + kb]; a.y = Arow[k + kb + 1];
    const float* Bp = B + (size_t)(k + kb) * Nn + n0 + mrow;
    v2f b0; b0.x = Bp[0];  b0.y = Bp[Nn];
    v2f b1; b1.x = Bp[16]; b1.y = Bp[Nn + 16];
    acc0 = __builtin_amdgcn_wmma_f32_16x16x4_f32(false, a, false, b0, (short)0, acc0, false, false);
    acc1 = __builtin_amdgcn_wmma_f32_16x16x4_f32(false, a, false, b1, (short)0, acc1, false, false);
  }
#else
  // scalar fallback (host pass / builtin missing): each lane owns rows r of D layout
  for (int r = 0; r < 8; ++r) {
    int row = m0 + r + ((lane >> 4) << 3);
    int c0 = n0 + (lane & 15);
    float s0 = 0.f, s1 = 0.f;
    for (int k = 0; k < K; ++k) {
      float av = A[(size_t)row * K + k];
      s0 += av * B[(size_t)k * Nn + c0];
      s1 += av * B[(size_t)k * Nn + c0 + 16];
    }
    acc0[r] = s0; acc1[r] = s1;
  }
#endif
  const int c0 = n0 + (lane & 15);
  const float bz0 = bias ? bias[c0] : 0.f;
  const float bz1 = bias ? bias[c0 + 16] : 0.f;
  for (int r = 0; r < 8; ++r) {
    int row = m0 + r + ((lane >> 4) << 3);
    float v0 = acc0[r] + bz0;
    float v1 = acc1[r] + bz1;
    float* Cp = C + (size_t)row * Nn + c0;
    if (rowscale) {
      float w = rowscale[(size_t)row * rs_stride];
      Cp[0]  += w * v0;
      Cp[16] += w * v1;
    } else {
      Cp[0]  = v0;
      Cp[16] = v1;
    }
  }
}

// ---------------- elementwise / small kernels ----------------
__global__ void prep_kernel(const float* __restrict__ ft, const float* __restrict__ fi,
                            float* __restrict__ xm) {
  int idx = blockIdx.x * blockDim.x + threadIdx.x;
  if (idx < NN * HH) xm[idx] = 0.5f * (ft[idx] + fi[idx]);
}

__global__ void mask_set_kernel(const int* __restrict__ midx, const float* __restrict__ mtok,
                                float* __restrict__ xm) {
  int idx = blockIdx.x * blockDim.x + threadIdx.x;
  if (idx >= NMC * HH) return;
  int row = idx / HH, c = idx % HH;
  xm[(size_t)midx[row] * HH + c] = mtok[c];
}

__global__ void gate_kernel(const float* __restrict__ xm, const float* __restrict__ gW,
                            const float* __restrict__ gb, float* __restrict__ wsel) {
  int n = blockIdx.x * blockDim.x + threadIdx.x;
  if (n >= NN) return;
  float logits[NEXPC];
  for (int e = 0; e < NEXPC; ++e) logits[e] = gb[e];
  const float* x = xm + (size_t)n * HH;
  for (int d = 0; d < HH; ++d) {
    float xv = x[d];
    const float* gr = gW + (size_t)d * NEXPC;
    for (int e = 0; e < NEXPC; ++e) logits[e] += xv * gr[e];
  }
  float mx = logits[0];
  for (int e = 1; e < NEXPC; ++e) mx = fmaxf(mx, logits[e]);
  float p[NEXPC]; float s = 0.f;
  for (int e = 0; e < NEXPC; ++e) { p[e] = expf(logits[e] - mx); s += p[e]; }
  for (int e = 0; e < NEXPC; ++e) p[e] /= s;
  int i1 = 0;
  for (int e = 1; e < NEXPC; ++e) if (p[e] > p[i1]) i1 = e;
  int i2 = (i1 == 0) ? 1 : 0;
  for (int e = 0; e < NEXPC; ++e) if (e != i1 && p[e] > p[i2]) i2 = e;
  float s2 = p[i1] + p[i2];
  for (int e = 0; e < NEXPC; ++e) wsel[n * NEXPC + e] = 0.f;
  wsel[n * NEXPC + i1] = p[i1] / s2;
  wsel[n * NEXPC + i2] = p[i2] / s2;
}

__global__ void zerof_kernel(float* __restrict__ p, int n) {
  int i = blockIdx.x * blockDim.x + threadIdx.x;
  if (i < n) p[i] = 0.f;
}

__global__ void fillu_kernel(unsigned* __restrict__ p, unsigned v, int n) {
  int i = blockIdx.x * blockDim.x + threadIdx.x;
  if (i < n) p[i] = v;
}

// In-place per-row LayerNorm(gamma,beta) + exact GELU. One block (256 thr) per row.
__global__ __launch_bounds__(256) void ln_gelu_kernel(float* __restrict__ Y,
                                                      const float* __restrict__ g,
                                                      const float* __restrict__ be) {
  float* y = Y + (size_t)blockIdx.x * HH;
  __shared__ float sh1[8], sh2[8], mv[2];
  int tid = threadIdx.x;
  float s1 = 0.f, s2 = 0.f;
  for (int c = tid; c < HH; c += 256) { float v = y[c]; s1 += v; s2 += v * v; }
  s1 = wred(s1); s2 = wred(s2);
  if ((tid & 31) == 0) { sh1[tid >> 5] = s1; sh2[tid >> 5] = s2; }
  __syncthreads();
  if (tid == 0) {
    float a = 0.f, b = 0.f;
    for (int i = 0; i < 8; ++i) { a += sh1[i]; b += sh2[i]; }
    float mean = a * (1.0f / HH);
    float var = fmaxf(b * (1.0f / HH) - mean * mean, 0.f);
    mv[0] = mean; mv[1] = rsqrtf(var + 1e-5f);
  }
  __syncthreads();
  float mean = mv[0], rstd = mv[1];
  for (int c = tid; c < HH; c += 256) {
    float v = (y[c] - mean) * rstd * g[c] + be[c];
    y[c] = gelu_f(v);
  }
}

// ---------------- GAT ----------------
__global__ __launch_bounds__(128) void gat_scores_kernel(
    const float* __restrict__ f, const float* __restrict__ al, const float* __restrict__ ar,
    float* __restrict__ el, float* __restrict__ er) {
  int gw = (blockIdx.x * blockDim.x + threadIdx.x) >> 5;
  int lane = threadIdx.x & 31;
  if (gw >= NN * HEADSC) return;
  int n = gw / HEADSC, hd = gw % HEADSC;
  const float* fr = f + (size_t)n * FDIM + hd * HH;
  const float* a = al + hd * HH;
  const float* b = ar + hd * HH;
  float sl = 0.f, sr = 0.f;
  for (int c = lane; c < HH; c += 32) { float v = fr[c]; sl += v * a[c]; sr += v * b[c]; }
  sl = wred(sl); sr = wred(sr);
  if (lane == 0) { el[gw] = sl; er[gw] = sr; }
}

__global__ void edge_max_kernel(const float* __restrict__ el, const float* __restrict__ er,
                                const int* __restrict__ src, const int* __restrict__ dst,
                                float* __restrict__ sbuf, unsigned* __restrict__ emaxu) {
  int idx = blockIdx.x * blockDim.x + threadIdx.x;
  if (idx >= EE * HEADSC) return;
  int eid = idx / HEADSC, hd = idx % HEADSC;
  float s = el[src[eid] * HEADSC + hd] + er[dst[eid] * HEADSC + hd];
  s = (s > 0.f) ? s : 0.2f * s;  // leaky_relu 0.2
  sbuf[idx] = s;
  atomicMax(&emaxu[dst[eid] * HEADSC + hd], fenc(s));
}

__global__ void fix_emax_kernel(const unsigned* __restrict__ emaxu, float* __restrict__ emaxf, int n) {
  int i = blockIdx.x * blockDim.x + threadIdx.x;
  if (i >= n) return;
  float d = fdec(emaxu[i]);
  emaxf[i] = (d > -1e38f && d < 1e38f) ? d : 0.f;  // where(isfinite, emax, 0)
}

__global__ void edge_exp_kernel(const float* __restrict__ sbuf, const float* __restrict__ emaxf,
                                const int* __restrict__ dst, float* __restrict__ exbuf,
                                float* __restrict__ den) {
  int idx = blockIdx.x * blockDim.x + threadIdx.x;
  if (idx >= EE * HEADSC) return;
  int eid = idx / HEADSC, hd = idx % HEADSC;
  float ex = expf(sbuf[idx] - emaxf[dst[eid] * HEADSC + hd]);
  exbuf[idx] = ex;
  atomicAdd(&den[dst[eid] * HEADSC + hd], ex);
}

// one wave per (edge, head)
__global__ __launch_bounds__(256) void edge_agg_kernel(
    const float* __restrict__ exbuf, const float* __restrict__ den,
    const int* __restrict__ src, const int* __restrict__ dst,
    const float* __restrict__ f, float* __restrict__ rst) {
  int gw = (blockIdx.x * (blockDim.x >> 5)) + (threadIdx.x >> 5);
  int lane = threadIdx.x & 31;
  if (gw >= EE * HEADSC) return;
  int eid = gw / HEADSC, hd = gw % HEADSC;
  int s = src[eid], d = dst[eid];
  float alpha = exbuf[gw] / fmaxf(den[d * HEADSC + hd], 1e-9f);
  const float* fs = f + (size_t)s * FDIM + hd * HH;
  float* rd = rst + (size_t)d * FDIM + hd * HH;
  for (int c = lane; c < HH; c += 32) atomicAdd(&rd[c], alpha * fs[c]);
}

__global__ void head_mean_kernel(const float* __restrict__ rst, const float* __restrict__ gb,
                                 float* __restrict__ h) {
  int idx = blockIdx.x * blockDim.x + threadIdx.x;
  if (idx >= NN * HH) return;
  int n = idx / HH, c = idx % HH;
  float s = 0.f;
  for (int hd = 0; hd < HEADSC; ++hd)
    s += rst[(size_t)n * FDIM + hd * HH + c] + gb[hd * HH + c];
  h[idx] = s * (1.0f / HEADSC);
}

// ---------------- decoder recon ----------------
__global__ void gather_hm_kernel(const float* __restrict__ h, const int* __restrict__ midx,
                                 float* __restrict__ hm) {
  int idx = blockIdx.x * blockDim.x + threadIdx.x;
  if (idx >= HM_PAD * HH) return;
  int row = idx / HH, c = idx % HH;
  hm[idx] = (row < NMC) ? h[(size_t)midx[row] * HH + c] : 0.f;
}

__global__ __launch_bounds__(128) void cos_kernel(const float* __restrict__ rec,
                                                  const float* __restrict__ feat,
                                                  const int* __restrict__ midx,
                                                  float* __restrict__ acc) {
  int gw = (blockIdx.x * blockDim.x + threadIdx.x) >> 5;
  int lane = threadIdx.x & 31;
  if (gw >= NMC) return;
  const float* r = rec + (size_t)gw * HH;
  const float* o = feat + (size_t)midx[gw] * HH;
  float d = 0.f, a = 0.f, b = 0.f;
  for (int c = lane; c < HH; c += 32) { float rv = r[c], ov = o[c]; d += rv * ov; a += rv * rv; b += ov * ov; }
  d = wred(d); a = wred(a); b = wred(b);
  if (lane == 0) {
    float cosv = d / fmaxf(sqrtf(a) * sqrtf(b), 1e-8f);
    float t = 1.0f - cosv;
    atomicAdd(acc, (t * t) * (1.0f / NMC));  // GAMMA = 2
  }
}

// ---------------- SPD pair loss: one wave per (i,j) ----------------
__global__ __launch_bounds__(256) void spd_kernel(
    const float* __restrict__ pa, const float* __restrict__ pb,
    const float* __restrict__ b1, const float* __restrict__ g, const float* __restrict__ be,
    const float* __restrict__ W2, const float* __restrict__ b2,
    const float* __restrict__ spd, float* __restrict__ acc) {
  const int lane = threadIdx.x & 31;
  const int wpb = blockDim.x >> 5;
  const int total = NN * NN;
  for (int p = blockIdx.x * wpb + (threadIdx.x >> 5); p < total; p += gridDim.x * wpb) {
    int i = p / NN, j = p % NN;
    const float* pai = pa + (size_t)i * HH;
    const float* pbj = pb + (size_t)j * HH;
    float v[HH / 32];
    float s1 = 0.f, s2 = 0.f;
#pragma unroll
    for (int t = 0; t < HH / 32; ++t) {
      int c = lane + 32 * t;
      float x = pai[c] + pbj[c] + b1[c];
      v[t] = x; s1 += x; s2 += x * x;
    }
    s1 = wred(s1); s2 = wred(s2);
    float mean = s1 * (1.0f / HH);
    float var = fmaxf(s2 * (1.0f / HH) - mean * mean, 0.f);
    float rstd = rsqrtf(var + 1e-5f);
    float dot = 0.f;
#pragma unroll
    for (int t = 0; t < HH / 32; ++t) {
      int c = lane + 32 * t;
      float z = (v[t] - mean) * rstd * g[c] + be[c];
      dot += gelu_f(z) * W2[c];
    }
    dot = wred(dot);
    if (lane == 0) {
      float diff = (dot + b2[0]) - spd[p];
      atomicAdd(acc, diff * diff);
    }
  }
}

__global__ void finalize_kernel(const float* __restrict__ acc, float* __restrict__ out) {
  if (threadIdx.x == 0 && blockIdx.x == 0)
    out[0] = acc[0] + 0.5f * (acc[1] / (float)(NN * NN));  // LAMBDA_SPD = 0.5
}

__global__ void copy_h_kernel(const float* __restrict__ h, float* __restrict__ out) {
  int i = blockIdx.x * blockDim.x + threadIdx.x;
  if (i < NN * HH) out[i] = h[i];
}

// ---------------- host driver ----------------
static inline void launch_gemm(const float* A, const float* B, const float* bias, float* C,
                               int M, int Nn, int K, const float* rowscale, int rs_stride,
                               hipStream_t s) {
  dim3 grid(Nn / 32, M / 16);
  gemm_wmma_f32<<<grid, dim3(32), 0, s>>>(A, B, bias, C, Nn, K, rowscale, rs_stride);
}

extern "C" void kernel_launch(void* const* d_in, const int* in_sizes, int n_in,
                              void* d_out, int out_size, void* d_ws, size_t ws_size,
                              hipStream_t stream) {
  (void)in_sizes; (void)n_in; (void)out_size; (void)ws_size;
  const float* feat_text  = (const float*)d_in[0];
  const float* feat_image = (const float*)d_in[1];
  const float* spd_matrix = (const float*)d_in[2];
  const int*   src        = (const int*)d_in[3];
  const int*   dst        = (const int*)d_in[4];
  const int*   mask_idx   = (const int*)d_in[5];
  const float* mask_token = (const float*)d_in[6];
  const float* gate_W     = (const float*)d_in[7];
  const float* gate_b     = (const float*)d_in[8];
  const float* exp_W1     = (const float*)d_in[9];
  const float* exp_b1     = (const float*)d_in[10];
  const float* exp_g1     = (const float*)d_in[11];
  const float* exp_be1    = (const float*)d_in[12];
  const float* exp_W2     = (const float*)d_in[13];
  const float* exp_b2     = (const float*)d_in[14];
  const float* gat_W      = (const float*)d_in[15];
  const float* gat_al     = (const float*)d_in[16];
  const float* gat_ar     = (const float*)d_in[17];
  const float* gat_b      = (const float*)d_in[18];
  const float* dec_W1     = (const float*)d_in[19];
  const float* dec_b1     = (const float*)d_in[20];
  const float* dec_g      = (const float*)d_in[21];
  const float* dec_be     = (const float*)d_in[22];
  const float* dec_W2     = (const float*)d_in[23];
  const float* dec_b2     = (const float*)d_in[24];
  const float* spd_W1     = (const float*)d_in[25];
  const float* spd_b1     = (const float*)d_in[26];
  const float* spd_g      = (const float*)d_in[27];
  const float* spd_be     = (const float*)d_in[28];
  const float* spd_W2     = (const float*)d_in[29];
  const float* spd_b2     = (const float*)d_in[30];
  float* out = (float*)d_out;

  // workspace carve-up (floats)
  float* w = (float*)d_ws;
  float* xm    = w; w += NN * HH;
  float* wsel  = w; w += NN * NEXPC;
  float* eh    = w; w += NN * HH;
  float* hbuf  = w; w += NN * HH;
  float* fbuf  = w; w += NN * FDIM;
  float* rst   = w; w += NN * FDIM;
  float* el    = w; w += NN * HEADSC;
  float* er    = w; w += NN * HEADSC;
  float* emaxf = w; w += NN * HEADSC;
  unsigned* emaxu = (unsigned*)w; w += NN * HEADSC;
  float* sbuf  = w; w += EE * HEADSC;
  float* exbuf = w; w += EE * HEADSC;
  float* den   = w; w += NN * HEADSC;
  float* hm    = w; w += HM_PAD * HH;
  float* z48   = w; w += HM_PAD * HH;
  float* rec48 = w; w += HM_PAD * HH;
  float* pa    = w; w += NN * HH;
  float* pb    = w; w += NN * HH;
  float* acc   = w; w += 2;

  // ---- prep + masking + gating ----
  prep_kernel<<<(NN * HH + 255) / 256, 256, 0, stream>>>(feat_text, feat_image, xm);
  mask_set_kernel<<<(NMC * HH + 255) / 256, 256, 0, stream>>>(mask_idx, mask_token, xm);
  gate_kernel<<<(NN + 127) / 128, 128, 0, stream>>>(xm, gate_W, gate_b, wsel);

  // ---- MoE: dense experts, gate-weighted accumulate into hbuf ----
  zerof_kernel<<<(NN * HH + 255) / 256, 256, 0, stream>>>(hbuf, NN * HH);
  for (int e = 0; e < NEXPC; ++e) {
    launch_gemm(xm, exp_W1 + (size_t)e * HH * HH, exp_b1 + e * HH, eh, NN, HH, HH, nullptr, 0, stream);
    ln_gelu_kernel<<<NN, 256, 0, stream>>>(eh, exp_g1 + e * HH, exp_be1 + e * HH);
    launch_gemm(eh, exp_W2 + (size_t)e * HH * HH, exp_b2 + e * HH, hbuf, NN, HH, HH, wsel + e, NEXPC, stream);
  }

  // ---- GAT layers ----
  for (int l = 0; l < LC; ++l) {
    launch_gemm(hbuf, gat_W + (size_t)l * HH * FDIM, nullptr, fbuf, NN, FDIM, HH, nullptr, 0, stream);
    gat_scores_kernel<<<(NN * HEADSC + 3) / 4, 128, 0, stream>>>(
        fbuf, gat_al + l * HEADSC * HH, gat_ar + l * HEADSC * HH, el, er);
    fillu_kernel<<<(NN * HEADSC + 255) / 256, 256, 0, stream>>>(emaxu, 0x007FFFFFu, NN * HEADSC); // enc(-inf)
    zerof_kernel<<<(NN * HEADSC + 255) / 256, 256, 0, stream>>>(den, NN * HEADSC);
    zerof_kernel<<<(NN * FDIM + 255) / 256, 256, 0, stream>>>(rst, NN * FDIM);
    edge_max_kernel<<<(EE * HEADSC + 255) / 256, 256, 0, stream>>>(el, er, src, dst, sbuf, emaxu);
    fix_emax_kernel<<<(NN * HEADSC + 255) / 256, 256, 0, stream>>>(emaxu, emaxf, NN * HEADSC);
    edge_exp_kernel<<<(EE * HEADSC + 255) / 256, 256, 0, stream>>>(sbuf, emaxf, dst, exbuf, den);
    edge_agg_kernel<<<(EE * HEADSC) / 8, 256, 0, stream>>>(exbuf, den, src, dst, fbuf, rst);
    head_mean_kernel<<<(NN * HH + 255) / 256, 256, 0, stream>>>(rst, gat_b + l * FDIM, hbuf);
  }

  // ---- losses ----
  zerof_kernel<<<1, 32, 0, stream>>>(acc, 2);
  gather_hm_kernel<<<(HM_PAD * HH + 255) / 256, 256, 0, stream>>>(hbuf, mask_idx, hm);
  for (int d = 0; d < 2; ++d) {
    launch_gemm(hm, dec_W1 + (size_t)d * HH * HH, dec_b1 + d * HH, z48, HM_PAD, HH, HH, nullptr, 0, stream);
    ln_gelu_kernel<<<HM_PAD, 256, 0, stream>>>(z48, dec_g + d * HH, dec_be + d * HH);
    launch_gemm(z48, dec_W2 + (size_t)d * HH * HH, dec_b2 + d * HH, rec48, HM_PAD, HH, HH, nullptr, 0, stream);
    cos_kernel<<<(NMC + 3) / 4, 128, 0, stream>>>(rec48, (d == 0) ? feat_text : feat_image, mask_idx, acc);
  }

  launch_gemm(hbuf, spd_W1, nullptr, pa, NN, HH, HH, nullptr, 0, stream);
  launch_gemm(hbuf, spd_W1 + (size_t)HH * HH, nullptr, pb, NN, HH, HH, nullptr, 0, stream);
  spd_kernel<<<2048, 256, 0, stream>>>(pa, pb, spd_b1, spd_g, spd_be, spd_W2, spd_b2, spd_matrix, acc + 1);

  finalize_kernel<<<1, 1, 0, stream>>>(acc, out);
  copy_h_kernel<<<(NN * HH + 255) / 256, 256, 0, stream>>>(hbuf, out + 1);
}